// GCN_6975026889094
// MI455X (gfx1250) — compile-verified
//
#include <hip/hip_runtime.h>
#include <hip/hip_bf16.h>

typedef float v2f __attribute__((ext_vector_type(2)));
typedef float v8f __attribute__((ext_vector_type(8)));

#define WAVE 32

// ---------------- degree + normalization ----------------
__global__ void deg_kernel(const int* __restrict__ src, const int* __restrict__ dst,
                           float* __restrict__ deg_out, float* __restrict__ deg_in, int E) {
  int i = blockIdx.x * blockDim.x + threadIdx.x;
  if (i < E) {
    atomicAdd(&deg_out[src[i]], 1.0f);
    atomicAdd(&deg_in[dst[i]], 1.0f);
  }
}

__global__ void cfac_kernel(const float* __restrict__ deg_out, const float* __restrict__ deg_in,
                            float* __restrict__ c_src, float* __restrict__ c_dst, int N) {
  int i = blockIdx.x * blockDim.x + threadIdx.x;
  if (i < N) {
    float a = deg_out[i];
    float b = deg_in[i];
    c_src[i] = (a > 0.0f) ? rsqrtf(a) : 0.0f;
    c_dst[i] = (b > 0.0f) ? rsqrtf(b) : 0.0f;
  }
}

// ---------------- WMMA f32 GEMM: Y = (X * c_src[:,None]) @ W ----------------
// Block handles a 16-row M tile; wave w handles N columns [16w, 16w+16).
// A tile staged in LDS (scaled by c_src), stride K+4 floats -> conflict-free b64 reads.
// VGPR layouts per CDNA5 ISA 7.12.2:
//   A (16x4 f32): lanes 0-15 row M=lane (K = k0,k0+1), lanes 16-31 row M=lane-16 (K = k0+2,k0+3)
//   B (4x16 f32): lanes 0-15 col N=lane (K = k0,k0+1), lanes 16-31 col N=lane-16 (K = k0+2,k0+3)
//   C/D (16x16): vgpr r, lanes 0-15 -> (M=r, N=lane); lanes 16-31 -> (M=r+8, N=lane-16)
template <int K, int NOUT>
__global__ __launch_bounds__(WAVE*(NOUT/16)) void gemm_wmma_kernel(
    const float* __restrict__ X, const float* __restrict__ W,
    const float* __restrict__ csrc, float* __restrict__ Y, int N) {
  constexpr int LSTRIDE = K + 4;
  constexpr int NT = WAVE * (NOUT / 16);
  __shared__ float As[16 * LSTRIDE];

  const int m0  = blockIdx.x * 16;
  const int tid = threadIdx.x;
  const bool full_tile = (m0 + 16 <= N);

  // cooperative A-tile load (float4), pre-scaled by c_src
  constexpr int NV = 16 * K / 4;
  if (full_tile) {
#pragma unroll
    for (int v = tid; v < NV; v += NT) {
      int row  = v / (K / 4);
      int col4 = (v % (K / 4)) * 4;
      int gr   = m0 + row;
      float4 x = *(const float4*)(X + (size_t)gr * K + col4);
      float  s = csrc[gr];
      float* p = As + row * LSTRIDE + col4;
      p[0] = x.x * s; p[1] = x.y * s; p[2] = x.z * s; p[3] = x.w * s;
    }
  } else {
    for (int v = tid; v < NV; v += NT) {
      int row  = v / (K / 4);
      int col4 = (v % (K / 4)) * 4;
      int gr   = m0 + row;
      float4 x = make_float4(0.f, 0.f, 0.f, 0.f);
      float  s = 0.f;
      if (gr < N) {
        x = *(const float4*)(X + (size_t)gr * K + col4);
        s = csrc[gr];
      }
      float* p = As + row * LSTRIDE + col4;
      p[0] = x.x * s; p[1] = x.y * s; p[2] = x.z * s; p[3] = x.w * s;
    }
  }
  __syncthreads();

  const int wv   = tid / WAVE;
  const int lane = tid % WAVE;
  const int n0   = wv * 16;
  const int half = (lane < 16) ? 0 : 2;
  const int r15  = lane & 15;

  v8f acc = {};
  const float* Arow = As + r15 * LSTRIDE;

#pragma unroll 4
  for (int k0 = 0; k0 < K; k0 += 4) {
    v2f a, b;
    a.x = Arow[k0 + half];
    a.y = Arow[k0 + half + 1];
    b.x = W[(size_t)(k0 + half) * NOUT + n0 + r15];
    b.y = W[(size_t)(k0 + half + 1) * NOUT + n0 + r15];
    acc = __builtin_amdgcn_wmma_f32_16x16x4_f32(false, a, false, b,
                                                (short)0, acc, false, false);
  }

  const int rowoff = (lane < 16) ? 0 : 8;
  float* Ybase = Y + (size_t)(m0 + rowoff) * NOUT + n0 + r15;
  if (full_tile) {
#pragma unroll
    for (int r = 0; r < 8; ++r) {
      Ybase[(size_t)r * NOUT] = acc[r];
    }
  } else {
#pragma unroll
    for (int r = 0; r < 8; ++r) {
      int gr = m0 + r + rowoff;
      if (gr < N) Y[(size_t)gr * NOUT + n0 + r15] = acc[r];
    }
  }
}

// ---------------- scatter-add SpMM (one wave per edge) ----------------
__global__ void scatter128(const float* __restrict__ Yin, const int* __restrict__ src,
                           const int* __restrict__ dst, float* __restrict__ agg, int E) {
  int t = blockIdx.x * blockDim.x + threadIdx.x;
  int e = t >> 5, lane = t & 31;
  if (e >= E) return;
  int s = src[e], d = dst[e];
  const float4 v = *(const float4*)(Yin + (size_t)s * 128 + lane * 4);
  float* p = agg + (size_t)d * 128 + lane * 4;
  atomicAdd(p + 0, v.x); atomicAdd(p + 1, v.y);
  atomicAdd(p + 2, v.z); atomicAdd(p + 3, v.w);
}

__global__ void scatter64(const float* __restrict__ Yin, const int* __restrict__ src,
                          const int* __restrict__ dst, float* __restrict__ agg, int E) {
  int t = blockIdx.x * blockDim.x + threadIdx.x;
  int e = t >> 5, lane = t & 31;
  if (e >= E) return;
  int s = src[e], d = dst[e];
  const float2 v = *(const float2*)(Yin + (size_t)s * 64 + lane * 2);
  float* p = agg + (size_t)d * 64 + lane * 2;
  atomicAdd(p + 0, v.x); atomicAdd(p + 1, v.y);
}

// ---------------- epilogues ----------------
__global__ void relu_bias_kernel(const float* __restrict__ agg, const float* __restrict__ c_dst,
                                 const float* __restrict__ b, float* __restrict__ h, int N) {
  size_t i = (size_t)blockIdx.x * blockDim.x + threadIdx.x;
  size_t total = (size_t)N * 128;
  if (i < total) {
    int node = (int)(i >> 7), j = (int)(i & 127);
    float v = agg[i] * c_dst[node] + b[j];
    h[i] = fmaxf(v, 0.0f);
  }
}

__global__ void finish_kernel(float* __restrict__ out, const float* __restrict__ c_dst,
                              const float* __restrict__ b, int N) {
  size_t i = (size_t)blockIdx.x * blockDim.x + threadIdx.x;
  size_t total = (size_t)N * 64;
  if (i < total) {
    int node = (int)(i >> 6), j = (int)(i & 63);
    out[i] = out[i] * c_dst[node] + b[j];
  }
}

// ---------------- launch ----------------
extern "C" void kernel_launch(void* const* d_in, const int* in_sizes, int n_in,
                              void* d_out, int out_size, void* d_ws, size_t ws_size,
                              hipStream_t stream) {
  const float* features = (const float*)d_in[0];
  const float* W1 = (const float*)d_in[1];
  const float* b1 = (const float*)d_in[2];
  const float* W2 = (const float*)d_in[3];
  const float* b2 = (const float*)d_in[4];
  const int*   src = (const int*)d_in[5];
  const int*   dst = (const int*)d_in[6];

  const int N = in_sizes[0] / 512;
  const int E = in_sizes[5];
  float* out = (float*)d_out;

  // workspace layout (floats)
  float* ws      = (float*)d_ws;
  float* deg_out = ws;
  float* deg_in  = deg_out + N;
  float* c_src   = deg_in + N;
  float* c_dst   = c_src + N;
  float* Y1      = c_dst + N;               // N*128 (reused as h after layer-1 epilogue)
  float* AGG1    = Y1 + (size_t)N * 128;    // N*128 (reused as Y2 in layer 2)

  // 1) degrees + normalization
  hipMemsetAsync(deg_out, 0, 2 * (size_t)N * sizeof(float), stream);
  deg_kernel<<<(E + 255) / 256, 256, 0, stream>>>(src, dst, deg_out, deg_in, E);
  cfac_kernel<<<(N + 255) / 256, 256, 0, stream>>>(deg_out, deg_in, c_src, c_dst, N);

  const int mt = (N + 15) / 16;

  // 2) layer 1: transform (WMMA), aggregate, relu+bias
  gemm_wmma_kernel<512, 128><<<mt, 256, 0, stream>>>(features, W1, c_src, Y1, N);
  hipMemsetAsync(AGG1, 0, (size_t)N * 128 * sizeof(float), stream);
  {
    long long tthreads = (long long)E * 32;
    scatter128<<<(int)((tthreads + 255) / 256), 256, 0, stream>>>(Y1, src, dst, AGG1, E);
  }
  {
    long long total = (long long)N * 128;
    relu_bias_kernel<<<(int)((total + 255) / 256), 256, 0, stream>>>(AGG1, c_dst, b1, Y1, N);
  }

  // 3) layer 2: transform (WMMA), aggregate into d_out, scale+bias in place
  gemm_wmma_kernel<128, 64><<<mt, 128, 0, stream>>>(Y1, W2, c_src, AGG1, N);
  hipMemsetAsync(out, 0, (size_t)N * 64 * sizeof(float), stream);
  {
    long long tthreads = (long long)E * 32;
    scatter64<<<(int)((tthreads + 255) / 256), 256, 0, stream>>>(AGG1, src, dst, out, E);
  }
  {
    long long total = (long long)N * 64;
    finish_kernel<<<(int)((total + 255) / 256), 256, 0, stream>>>(out, c_dst, b2, N);
  }
}